// DecodeBoxLayer_79413945303925
// MI455X (gfx1250) — compile-verified
//
#include <hip/hip_runtime.h>
#include <hip/hip_bf16.h>

// DecodeBoxLayer: boxes [64, 100000, 4] int32 (y,x,h,w) ->
// corners [64, 100000, 4, 2] int32.
// Pure streaming transform: 102.4 MB in, 204.8 MB out => HBM-bandwidth bound
// (~13 us at 23.3 TB/s). Strategy: 1 box per lane, b128 load + 2x b128 store,
// all with non-temporal cache hints so the one-touch streams don't thrash the
// 192 MB L2.

typedef int v4i __attribute__((ext_vector_type(4)));

__global__ __launch_bounds__(256) void decode_box_corners_kernel(
    const v4i* __restrict__ boxes,   // one int4 per box: (y, x, h, w)
    v4i* __restrict__ out,           // two int4 per box: (x1,y1,x2,y1),(x2,y2,x1,y2)
    int n_boxes) {
  int i = blockIdx.x * blockDim.x + threadIdx.x;
  if (i >= n_boxes) return;

  // global_load_b128 th:TH_NT — box read exactly once, bypass temporal caching.
  v4i b = __builtin_nontemporal_load(&boxes[i]);

  const int y = b.x;
  const int x = b.y;
  const int h = b.z;
  const int w = b.w;

  const int x1 = x << 1;
  const int x2 = (x + w) << 1;
  const int y1 = y << 1;
  const int y2 = (y + h) << 1;

  v4i c0, c1;
  c0.x = x1; c0.y = y1; c0.z = x2; c0.w = y1;   // corners 0,1
  c1.x = x2; c1.y = y2; c1.z = x1; c1.w = y2;   // corners 2,3

  // Two global_store_b128 th:TH_NT covering the contiguous 32B output record.
  // Lanes of a wave cover a contiguous 1 KB span between the two stores, so
  // cachelines are fully written and rinse straight to HBM.
  __builtin_nontemporal_store(c0, &out[2 * i + 0]);
  __builtin_nontemporal_store(c1, &out[2 * i + 1]);
}

extern "C" void kernel_launch(void* const* d_in, const int* in_sizes, int n_in,
                              void* d_out, int out_size, void* d_ws, size_t ws_size,
                              hipStream_t stream) {
  (void)n_in; (void)out_size; (void)d_ws; (void)ws_size;

  const v4i* boxes = (const v4i*)d_in[0];   // int32 input
  v4i* out = (v4i*)d_out;                   // int32 output (corners)

  const int n_boxes = in_sizes[0] / 4;      // 64 * 100000 = 6,400,000
  const int block = 256;                    // 8 waves (wave32) per block
  const int grid = (n_boxes + block - 1) / block;

  decode_box_corners_kernel<<<grid, block, 0, stream>>>(boxes, out, n_boxes);
}